// UnfiltBackproj3DBatch_6416681140697
// MI455X (gfx1250) — compile-verified
//
#include <hip/hip_runtime.h>

// ---------------------------------------------------------------------------
// out[b,0,i*16+p, j*4+q, l*4+r] = sum_k y[b,k,i,j,l] * W[k,0,p,q,r]
// GEMM view: C[m,n] = sum_k A[m,k] B[k,n]
//   m = b*4096 + i*64 + j*8 + l   (M = 16384)
//   n = p*16 + q*4 + r            (N = 256)
//   K = 64
// y element (m,k)  -> y[b*262144 + k*4096 + (m & 4095)]
// W element (k,n)  -> W[k*256 + n]
// out element(m,n) -> out[b*1048576 + i*16384 + p*1024 + j*128 + q*32 + l*4 + r]
//
// Roofline (MI455X): 0.54 GFLOP vs ~21 MB HBM traffic -> memory-bound floor
// ~0.9 us @ 23.3 TB/s. V_WMMA_F32_16X16X4_F32 keeps exact fp32 semantics and
// lifts matrix throughput ~8x over VALU FMA so compute sits at/below the
// memory floor. All 21 MB fits the 192 MB L2, so A-refetch across N-tiles is
// L2-resident; one 16x16 tile per wave maximizes occupancy (16384 waves).
// Stores staged through LDS so every global store is a 128B coalesced segment.
// ---------------------------------------------------------------------------

typedef __attribute__((ext_vector_type(2))) float v2f;
typedef __attribute__((ext_vector_type(8))) float v8f;

#define WAVES_PER_BLOCK 8

__global__ __launch_bounds__(256)
void backproj_wmma_f32_kernel(const float* __restrict__ y,
                              const float* __restrict__ W,
                              float* __restrict__ out)
{
    __shared__ float tile[WAVES_PER_BLOCK][16][17];   // +1 pad vs 64 banks

    const int tid  = threadIdx.x;
    const int lane = tid & 31;
    const int wid  = tid >> 5;

    const int w     = blockIdx.x * WAVES_PER_BLOCK + wid;  // global wave id
    const int ntile = w & 15;                              // 16 N tiles
    const int mtile = w >> 4;                              // 1024 M tiles
    const int n0    = ntile << 4;
    const int m0    = mtile << 4;                          // mult of 16 -> l0=0, j0 even

    // decompose m0: m = b*4096 + i*64 + j*8 + l  (all 16 rows share b and i)
    const int  b  = m0 >> 12;
    const int  s0 = m0 & 4095;
    const long ybase = (long)b * 262144 + s0;

    const int half = lane >> 4;      // 0 -> K pair (k0,k0+1), 1 -> (k0+2,k0+3)
    const int l15  = lane & 15;

    // A: this lane's row address (add k*4096 per step)
    const long aAddr = ybase + l15;
    // B: this lane's column (add k*256 per step)
    const int  bCol  = n0 + l15;

    v8f acc = {0.f, 0.f, 0.f, 0.f, 0.f, 0.f, 0.f, 0.f};

    #pragma unroll
    for (int k0 = 0; k0 < 64; k0 += 4) {
        const int kk = k0 + (half << 1);
        v2f a, bm;
        // A-matrix 16x4 f32 layout: VGPR0=K(kk), VGPR1=K(kk+1); upper lanes kk+2/kk+3
        a.x  = y[aAddr + (long)kk * 4096];
        a.y  = y[aAddr + (long)(kk + 1) * 4096];
        // B-matrix 4x16 f32 layout: rows striped across lanes, mirrored K split
        bm.x = W[kk * 256 + bCol];
        bm.y = W[(kk + 1) * 256 + bCol];
        acc = __builtin_amdgcn_wmma_f32_16x16x4_f32(
                  /*neg_a=*/false, a, /*neg_b=*/false, bm,
                  /*c_mod=*/(short)0, acc, /*reuse_a=*/false, /*reuse_b=*/false);
    }

    // C/D layout: VGPR g -> (M=g, N=lane) for lanes 0-15, (M=g+8, N=lane-16) for 16-31
    const int rbase = half << 3;
    #pragma unroll
    for (int g = 0; g < 8; ++g)
        tile[wid][rbase + g][l15] = acc[g];

    // intra-wave LDS write -> cross-lane read ordering (CDNA5 split counters)
    asm volatile("s_wait_dscnt 0" ::: "memory");

    // Coalesced scatter. Tile covers dj in {0,1} (j=j0+dj), l=0..7, fixed p,
    // q=0..3, r=0..3. For each (dj,q) the 32 outputs (l*4+r) are contiguous:
    // lane id == l*4+r -> one 128B fully-coalesced store per segment.
    const int  i  = s0 >> 6;
    const int  j0 = (s0 >> 3) & 7;
    const int  p  = n0 >> 4;
    const int  el = lane >> 2;   // l
    const int  er = lane & 3;    // r
    const long obase = (long)b * 1048576 + (long)i * 16384
                     + (long)p * 1024 + (long)j0 * 128;

    #pragma unroll
    for (int dj = 0; dj < 2; ++dj) {
        #pragma unroll
        for (int q = 0; q < 4; ++q) {
            const float v = tile[wid][dj * 8 + el][q * 4 + er];
            out[obase + dj * 128 + q * 32 + lane] = v;
        }
    }
}

extern "C" void kernel_launch(void* const* d_in, const int* in_sizes, int n_in,
                              void* d_out, int out_size, void* d_ws, size_t ws_size,
                              hipStream_t stream) {
    const float* y = (const float*)d_in[0];   // (4, 64, 64, 8, 8) fp32
    const float* W = (const float*)d_in[1];   // (64, 1, 16, 4, 4) fp32
    float* out = (float*)d_out;               // (4, 1, 1024, 32, 32) fp32

    // 16384 M-rows/16 * 256 N-cols/16 = 16384 tiles = 16384 waves
    // 8 waves (256 threads) per block -> 2048 blocks
    dim3 grid(2048), block(256);
    hipLaunchKernelGGL(backproj_wmma_f32_kernel, grid, block, 0, stream,
                       y, W, out);
}